// WorkingMemory_68066641707188
// MI455X (gfx1250) — compile-verified
//
#include <hip/hip_runtime.h>
#include <math.h>

typedef __attribute__((ext_vector_type(2))) float v2f;
typedef __attribute__((ext_vector_type(8))) float v8f;

#define B_      256
#define N_      65536
#define D_      512
#define TOPK    16
#define NORMEPS 1e-4f   // per-factor floor; product floor == reference's 1e-8
#define NCHUNKS 64
#define CHUNK   1024
#define QS      516   // LDS row stride in floats: 516 % 64 == 4 -> conflict-free b64 reads

// workspace layout (float offsets)
#define Q_OFF   0
#define QI_OFF  (Q_OFF + B_*D_)          // 131072 : qinv[b] = 1/||q_b||
#define KS_OFF  (QI_OFF + B_)            // 131328 : kscale[n] = imp[n]/||k_n||
#define PV_CNT  (16*NCHUNKS*2*16*TOPK)   // 524288
#define PV_OFF  (KS_OFF + N_)            // 196864
#define PI_OFF  (PV_OFF + PV_CNT)        // 721152

// Branchless sorted (descending) top-16 insert: bubbles candidate through.
__device__ __forceinline__ void topk_insert(float (&tv)[16], int (&ti)[16], float v, int idx) {
#pragma unroll
    for (int j = 0; j < 16; ++j) {
        const bool gt = v > tv[j];
        const float ov = tv[j]; const int oi = ti[j];
        tv[j] = gt ? v : ov;
        ti[j] = gt ? idx : oi;
        v  = gt ? ov : v;
        idx = gt ? oi : idx;
    }
}

// ---------------- Kernel 1: q = query @ Wq^T + bq, and qinv = 1/||q|| ----------------
__global__ __launch_bounds__(256)
void proj_kernel(const float* __restrict__ query, const float* __restrict__ Wq,
                 const float* __restrict__ bq, float* __restrict__ qbuf,
                 float* __restrict__ qinv)
{
    __shared__ float qrow[D_];
    __shared__ float red[256];
    const int b = blockIdx.x;
    const int tid = threadIdx.x;
    for (int i = tid; i < D_; i += 256) qrow[i] = query[(size_t)b*D_ + i];
    __syncthreads();
    float ssq = 0.f;
#pragma unroll
    for (int rep = 0; rep < 2; ++rep) {
        const int d = tid + rep*256;
        const float* wp = Wq + (size_t)d*D_;
        float acc = 0.f;
        for (int k = 0; k < D_; k += 4) {
            const float4 w4 = *reinterpret_cast<const float4*>(wp + k);
            acc += qrow[k]*w4.x + qrow[k+1]*w4.y + qrow[k+2]*w4.z + qrow[k+3]*w4.w;
        }
        acc += bq[d];
        qbuf[(size_t)b*D_ + d] = acc;
        ssq += acc*acc;
    }
    red[tid] = ssq;
    __syncthreads();
    for (int s = 128; s > 0; s >>= 1) {
        if (tid < s) red[tid] += red[tid+s];
        __syncthreads();
    }
    if (tid == 0) qinv[b] = 1.f / fmaxf(sqrtf(red[0]), NORMEPS);
}

// ------------- Kernel 2: kscale[n] = importance[n]/||mem_keys[n]|| (one wave32/key) -------------
__global__ __launch_bounds__(256)
void kscale_kernel(const float* __restrict__ mem_keys,
                   const float* __restrict__ importance,
                   float* __restrict__ kscale)
{
    const int wave = threadIdx.x >> 5;
    const int lane = threadIdx.x & 31;
    const int n = blockIdx.x*8 + wave;
    const float* kp = mem_keys + (size_t)n*D_ + lane*16;
    float ssq = 0.f;
#pragma unroll
    for (int i = 0; i < 16; i += 4) {
        const float4 v = *reinterpret_cast<const float4*>(kp + i);
        ssq += v.x*v.x + v.y*v.y + v.z*v.z + v.w*v.w;
    }
#pragma unroll
    for (int m = 16; m > 0; m >>= 1) ssq += __shfl_xor(ssq, m, 32);
    if (lane == 0) kscale[n] = importance[n] / fmaxf(sqrtf(ssq), NORMEPS);
}

// ---------------- Kernel 3: fp32 WMMA GEMM (q x K^T) + scale + running top-16 ----------------
// grid: (16 chunk-groups, 16 batch tiles), block: 128 threads (4 waves).
// Each wave owns chunk = blockIdx.x*4 + wave (1024 columns = 64 tiles of 16).
__global__ __launch_bounds__(128)
void gemm_topk_kernel(const float* __restrict__ mem_keys,
                      const float* __restrict__ qbuf,
                      const float* __restrict__ qinv,
                      const float* __restrict__ kscale,
                      float* __restrict__ pv, int* __restrict__ pi)
{
    __shared__ float qtile[16*QS];       // 16 x 512 q-tile, padded rows
    __shared__ float stile[4][16*17];    // per-wave score staging
    const int tid  = threadIdx.x;
    const int wave = tid >> 5;
    const int lane = tid & 31;
    const int bt   = blockIdx.y;
    const int m0   = bt * 16;
    const int chunk = blockIdx.x*4 + wave;
    const int nbase = chunk * CHUNK;

    // Stage 16x512 q tile in LDS (coalesced float4).
    for (int i = tid*4; i < 16*D_; i += 128*4) {
        const int row = i >> 9, colq = i & (D_-1);
        const float4 v = *reinterpret_cast<const float4*>(qbuf + (size_t)(m0+row)*D_ + colq);
        *reinterpret_cast<float4*>(&qtile[row*QS + colq]) = v;
    }
    __syncthreads();

    const int h   = lane >> 4;   // K-half for A/B operands, row-half for C
    const int col = lane & 15;   // N column within tile / A row (M)
    float qiv[8];
#pragma unroll
    for (int j = 0; j < 8; ++j) qiv[j] = qinv[m0 + j + 8*h];

    float tv[16]; int ti[16];
#pragma unroll
    for (int j = 0; j < 16; ++j) { tv[j] = -3.0e38f; ti[j] = 0; }

    float* smy = stile[wave];
    const float* ap = &qtile[(size_t)col*QS + 2*h];   // A: q row M=col, K slice

    for (int t = 0; t < CHUNK/16; ++t) {
        const int n0 = nbase + t*16;
        const int nk = n0 + col;
        const float* kp = mem_keys + (size_t)nk*D_ + 2*h;  // B: key row nk, K slice

        v8f c = {0.f,0.f,0.f,0.f,0.f,0.f,0.f,0.f};
#pragma unroll 8
        for (int k0 = 0; k0 < D_; k0 += 4) {
            const float2 av = *reinterpret_cast<const float2*>(ap + k0);
            const float2 bv = *reinterpret_cast<const float2*>(kp + k0);
            v2f aa; aa[0] = av.x; aa[1] = av.y;
            v2f bb; bb[0] = bv.x; bb[1] = bv.y;
            c = __builtin_amdgcn_wmma_f32_16x16x4_f32(
                    /*neg_a=*/false, aa, /*neg_b=*/false, bb,
                    /*c_mod=*/(short)0, c, /*reuse_a=*/false, /*reuse_b=*/false);
        }

        // Epilogue: scores = dots * qinv[row] * kscale[col]; stage tile to LDS.
        const float ks = kscale[nk];
#pragma unroll
        for (int j = 0; j < 8; ++j) {
            smy[(j + 8*h)*17 + col] = c[j] * (qiv[j] * ks);
        }
        __syncthreads();

        // Each lane owns (row = lane&15, column half = lane>>4): 8 candidates.
        const int r = lane & 15;
#pragma unroll
        for (int cc = 0; cc < 8; ++cc) {
            const int c2 = h*8 + cc;
            const float v = smy[r*17 + c2];
            if (v > tv[15]) topk_insert(tv, ti, v, n0 + c2);
        }
        __syncthreads();
    }

    // Write partial top-16 list: [bt][chunk][half][row][16]
    const size_t base = (((size_t)(bt*NCHUNKS + chunk)*2 + h)*16 + (lane & 15))*TOPK;
    float* pvp = pv + base;
    int*   pip = pi + base;
#pragma unroll
    for (int j = 0; j < 16; ++j) { pvp[j] = tv[j]; pip[j] = ti[j]; }
}

// ---------------- Kernel 4: merge partials -> top-16 -> softmax -> gather ----------------
// One wave32 per batch row.
__global__ __launch_bounds__(32)
void merge_kernel(const float* __restrict__ mem_values,
                  const float* __restrict__ pv, const int* __restrict__ pi,
                  float* __restrict__ out)
{
    __shared__ float lv[32*16];
    __shared__ int   li[32*16];
    __shared__ float lw[16];
    __shared__ int   lidx[16];
    const int b   = blockIdx.x;
    const int bt  = b >> 4;
    const int r   = b & 15;
    const int lane = threadIdx.x;

    float tv[16]; int ti[16];
#pragma unroll
    for (int j = 0; j < 16; ++j) { tv[j] = -3.0e38f; ti[j] = 0; }

    const int ncand = NCHUNKS*2*16;   // 2048 candidates for this row
    for (int i = lane; i < ncand; i += 32) {
        const int chunk = i >> 5;
        const int rem   = i & 31;
        const int h     = rem >> 4;
        const int j     = rem & 15;
        const size_t base = (((size_t)(bt*NCHUNKS + chunk)*2 + h)*16 + r)*TOPK + j;
        const float v = pv[base];
        if (v > tv[15]) topk_insert(tv, ti, v, pi[base]);
    }
#pragma unroll
    for (int j = 0; j < 16; ++j) { lv[lane*16+j] = tv[j]; li[lane*16+j] = ti[j]; }
    __syncthreads();

    if (lane == 0) {
        float ftv[16]; int fti[16];
#pragma unroll
        for (int j = 0; j < 16; ++j) { ftv[j] = -3.0e38f; fti[j] = 0; }
        for (int i = 0; i < 32*16; ++i) {
            const float v = lv[i];
            if (v > ftv[15]) topk_insert(ftv, fti, v, li[i]);
        }
        const float mx = ftv[0];   // sorted descending
        float e[16]; float s = 0.f;
#pragma unroll
        for (int j = 0; j < 16; ++j) { e[j] = expf(ftv[j]-mx); s += e[j]; }
        const float inv = 1.f/s;
#pragma unroll
        for (int j = 0; j < 16; ++j) { lw[j] = e[j]*inv; lidx[j] = fti[j]; }
    }
    __syncthreads();

    // Gather: each lane owns 16 consecutive dims.
    const int d0 = lane*16;
    float acc[16];
#pragma unroll
    for (int i = 0; i < 16; ++i) acc[i] = 0.f;
    for (int k = 0; k < 16; ++k) {
        const float w = lw[k];
        const float* vp = mem_values + (size_t)lidx[k]*D_ + d0;
#pragma unroll
        for (int i = 0; i < 16; i += 4) {
            const float4 v4 = *reinterpret_cast<const float4*>(vp + i);
            acc[i]   += w*v4.x; acc[i+1] += w*v4.y;
            acc[i+2] += w*v4.z; acc[i+3] += w*v4.w;
        }
    }
    float* op = out + (size_t)b*D_ + d0;
#pragma unroll
    for (int i = 0; i < 16; ++i) op[i] = acc[i];
}

extern "C" void kernel_launch(void* const* d_in, const int* in_sizes, int n_in,
                              void* d_out, int out_size, void* d_ws, size_t ws_size,
                              hipStream_t stream) {
    (void)in_sizes; (void)n_in; (void)out_size; (void)ws_size;
    const float* query      = (const float*)d_in[0];
    const float* mem_keys   = (const float*)d_in[1];
    const float* mem_values = (const float*)d_in[2];
    const float* importance = (const float*)d_in[3];
    const float* Wq         = (const float*)d_in[4];
    const float* bq         = (const float*)d_in[5];
    float* out = (float*)d_out;

    float* ws     = (float*)d_ws;
    float* qbuf   = ws + Q_OFF;
    float* qinv   = ws + QI_OFF;
    float* kscale = ws + KS_OFF;
    float* pv     = ws + PV_OFF;
    int*   pi     = (int*)(ws + PI_OFF);

    proj_kernel<<<B_, 256, 0, stream>>>(query, Wq, bq, qbuf, qinv);
    kscale_kernel<<<N_/8, 256, 0, stream>>>(mem_keys, importance, kscale);
    gemm_topk_kernel<<<dim3(16,16), 128, 0, stream>>>(mem_keys, qbuf, qinv, kscale, pv, pi);
    merge_kernel<<<B_, 32, 0, stream>>>(mem_values, pv, pi, out);
}